// GConv_18657337934238
// MI455X (gfx1250) — compile-verified
//
#include <hip/hip_runtime.h>

typedef float v2f __attribute__((ext_vector_type(2)));
typedef float v8f __attribute__((ext_vector_type(8)));
typedef int   i4v __attribute__((vector_size(4 * sizeof(int))));

#define AS1 __attribute__((address_space(1)))
#define AS3 __attribute__((address_space(3)))

#define N_OP     50000
#define N_DEV    5000
#define D_FEAT   128
#define DE       16
#define KDIM     144        // 128 + 16, = 36 * 4
#define OUT_F    128
#define LDSTRIDE 148        // padded row stride (floats): LDS-bank friendly, keeps 16B align per row
#define TPB_TILES 8         // edge tiles (of 16 edges) per block

#if __has_builtin(__builtin_amdgcn_global_load_async_to_lds_b128) && \
    __has_builtin(__builtin_amdgcn_global_load_async_to_lds_b32)  && \
    __has_builtin(__builtin_amdgcn_s_wait_asynccnt)
#define HAS_ASYNC 1
#else
#define HAS_ASYNC 0
#endif

// ---------------------------------------------------------------------------
// Stage one 16x144 A tile (16 edges: gathered src feats ++ edge feats) into an
// LDS buffer. Async path: 3 ASYNC ops per wave (2x b128 feats + 1x b32 ef).
// ---------------------------------------------------------------------------
__device__ __forceinline__ void prefetch_tile(
    const float* __restrict__ src_feats,
    const float* __restrict__ ef,
    const int*   __restrict__ src,
    const int*   __restrict__ dst,
    float* ldsBuf, int* ldsDstBuf,
    int ebase, int r, int cIdx)
{
    const int e  = ebase + r;
    const int sn = src[e];
    const float* g = src_feats + (size_t)sn * D_FEAT + cIdx * 8;
    float*       l = ldsBuf + r * LDSTRIDE + cIdx * 8;
#if HAS_ASYNC
    __builtin_amdgcn_global_load_async_to_lds_b128(
        (AS1 i4v*)g, (AS3 i4v*)l, 0, 0);
    __builtin_amdgcn_global_load_async_to_lds_b128(
        (AS1 i4v*)(g + 4), (AS3 i4v*)(l + 4), 0, 0);
    __builtin_amdgcn_global_load_async_to_lds_b32(
        (AS1 int*)(ef + (size_t)e * DE + cIdx),
        (AS3 int*)(ldsBuf + r * LDSTRIDE + D_FEAT + cIdx), 0, 0);
#else
    const float4* sp = (const float4*)g;
    float4 f0 = sp[0];
    float4 f1 = sp[1];
    float4* lp = (float4*)l;
    lp[0] = f0;
    lp[1] = f1;
    ldsBuf[r * LDSTRIDE + D_FEAT + cIdx] = ef[(size_t)e * DE + cIdx];
#endif
    if (cIdx == 0) ldsDstBuf[r] = dst[e];
}

// ---------------------------------------------------------------------------
// Per-edge-type fused kernel:
//   m[e,:] = concat(src_feats[src[e]], ef[e]) @ W + b      (WMMA f32 16x16x4)
//   sums[dst[e],:] += m[e,:] ;  cnt[dst[e]] += 1
// Double-buffered LDS staging via GLOBAL_LOAD_ASYNC_TO_LDS.
// ---------------------------------------------------------------------------
__global__ __launch_bounds__(256) void edge_conv_wmma(
    const float* __restrict__ src_feats,   // [n_src,128]
    const float* __restrict__ ef,          // [E,16]
    const int*   __restrict__ src,         // [E]
    const int*   __restrict__ dst,         // [E]
    const float* __restrict__ W,           // [144,128] row-major
    const float* __restrict__ bias,        // [128]
    float*       __restrict__ sums,        // [n_dst,128]
    float*       __restrict__ cnt,         // [n_dst]
    int ntiles)
{
    __shared__ float ldsA[2][16 * LDSTRIDE];
    __shared__ int   ldsDst[2][16];

    const int t    = threadIdx.x;
    const int lane = t & 31;           // wave32
    const int wave = t >> 5;           // 0..7 -> N-tile
    const int col0 = wave * 16;
    const int nsub = lane & 15;        // col within N-tile / row within M-tile
    const int half = lane >> 4;        // 0/1 (K-pair select per ISA layout)

    // ---- Load this wave's B fragments (W columns col0..col0+15) into VGPRs.
    // B 4x16 f32 layout: vgpr0 = rows {k0, k0+2}, vgpr1 = rows {k0+1, k0+3}
    // -> lane L holds W[k0 + 2*half + {0,1}][col0 + (L&15)].
    v2f Bw[36];
    #pragma unroll
    for (int s = 0; s < 36; ++s) {
        const int kk = 4 * s + 2 * half;
        Bw[s].x = W[(size_t)(kk + 0) * OUT_F + col0 + nsub];
        Bw[s].y = W[(size_t)(kk + 1) * OUT_F + col0 + nsub];
    }
    const float bval = bias[col0 + nsub];

    const int tile0 = blockIdx.x * TPB_TILES;
    const int tile1 = min(tile0 + TPB_TILES, ntiles);
    const int nt    = tile1 - tile0;
    if (nt <= 0) return;

    const int r    = t >> 4;   // 0..15: A-tile row this thread helps stage
    const int cIdx = t & 15;   // 0..15

    // Prime the pipeline: stage tile0 into buffer 0.
    prefetch_tile(src_feats, ef, src, dst, ldsA[0], ldsDst[0], tile0 * 16, r, cIdx);

    for (int i = 0; i < nt; ++i) {
        const int buf      = i & 1;
        const bool hasNext = (i + 1) < nt;

        // Issue next tile's staging into the other buffer (overlaps with WMMA).
        if (hasNext) {
            prefetch_tile(src_feats, ef, src, dst, ldsA[buf ^ 1], ldsDst[buf ^ 1],
                          (tile0 + i + 1) * 16, r, cIdx);
        }
#if HAS_ASYNC
        // Drain this buffer's 3 async ops; next buffer's 3 may remain in flight.
        if (hasNext) __builtin_amdgcn_s_wait_asynccnt(3);
        else         __builtin_amdgcn_s_wait_asynccnt(0);
#endif
        __syncthreads();   // all waves' staging of `buf` now visible

        // ---- 36 chained f32 WMMAs: C[16x16] += A[16x4] * B[4x16]
        // A 16x4 f32 layout: lane L<16 holds A[M=L][k0+{0,1}], L>=16 holds A[M=L-16][k0+{2,3}]
        v8f acc = {0.f, 0.f, 0.f, 0.f, 0.f, 0.f, 0.f, 0.f};
        #pragma unroll
        for (int s = 0; s < 36; ++s) {
            const int c = 4 * s + 2 * half;
            v2f a = *(const v2f*)(&ldsA[buf][nsub * LDSTRIDE + c]);
            acc = __builtin_amdgcn_wmma_f32_16x16x4_f32(
                false, a, false, Bw[s], (short)0, acc, false, false);
        }

        // ---- Scatter-add (+bias) into per-etype sum buffer.
        // C/D layout: vgpr j, lane L -> row M = j + 8*half, col N = L&15
        #pragma unroll
        for (int j = 0; j < 8; ++j) {
            const int m = j + half * 8;
            const int d = ldsDst[buf][m];
            unsafeAtomicAdd(&sums[(size_t)d * OUT_F + col0 + nsub], acc[j] + bval);
        }
        if (wave == 0 && lane < 16) {
            unsafeAtomicAdd(&cnt[ldsDst[buf][lane]], 1.0f);
        }
        __syncthreads();   // protect `buf` from the prefetch issued at i+1 (tile i+2)
    }
}

// ---------------------------------------------------------------------------
// Finalize: per-etype mean (0 for 0-degree), average across etypes, ReLU,
// concat with input features.
// ---------------------------------------------------------------------------
__global__ __launch_bounds__(256) void finalize3(
    const float* __restrict__ feats,
    const float* __restrict__ s0, const float* __restrict__ c0,
    const float* __restrict__ s1, const float* __restrict__ c1,
    const float* __restrict__ s2, const float* __restrict__ c2,
    float* __restrict__ out, int nnodes)
{
    const int t = blockIdx.x * blockDim.x + threadIdx.x;
    if (t >= nnodes * D_FEAT) return;
    const int node = t >> 7, c = t & 127;
    float m0 = 0.f, m1 = 0.f, m2 = 0.f;
    const float k0 = c0[node]; if (k0 > 0.f) m0 = s0[t] / k0;
    const float k1 = c1[node]; if (k1 > 0.f) m1 = s1[t] / k1;
    const float k2 = c2[node]; if (k2 > 0.f) m2 = s2[t] / k2;
    float v = (m0 + m1 + m2) * (1.0f / 3.0f);
    v = v > 0.f ? v : 0.f;
    out[(size_t)node * 256 + c]       = feats[t];
    out[(size_t)node * 256 + 128 + c] = v;
}

__global__ __launch_bounds__(256) void finalize2(
    const float* __restrict__ feats,
    const float* __restrict__ s0, const float* __restrict__ c0,
    const float* __restrict__ s1, const float* __restrict__ c1,
    float* __restrict__ out, int nnodes)
{
    const int t = blockIdx.x * blockDim.x + threadIdx.x;
    if (t >= nnodes * D_FEAT) return;
    const int node = t >> 7, c = t & 127;
    float m0 = 0.f, m1 = 0.f;
    const float k0 = c0[node]; if (k0 > 0.f) m0 = s0[t] / k0;
    const float k1 = c1[node]; if (k1 > 0.f) m1 = s1[t] / k1;
    float v = (m0 + m1) * 0.5f;
    v = v > 0.f ? v : 0.f;
    out[(size_t)node * 256 + c]       = feats[t];
    out[(size_t)node * 256 + 128 + c] = v;
}

// ---------------------------------------------------------------------------
extern "C" void kernel_launch(void* const* d_in, const int* in_sizes, int n_in,
                              void* d_out, int out_size, void* d_ws, size_t ws_size,
                              hipStream_t stream)
{
    const float* op_feats  = (const float*)d_in[0];
    const float* dev_feats = (const float*)d_in[1];

    // ETYPES order: link, prev, succ, place, serve; each contributes 5 inputs
    // (ef, src, dst, W, b) starting at index 2.
    const float* ef_link  = (const float*)d_in[2];
    const int*   src_link = (const int*)  d_in[3];
    const int*   dst_link = (const int*)  d_in[4];
    const float* W_link   = (const float*)d_in[5];
    const float* b_link   = (const float*)d_in[6];

    const float* ef_prev  = (const float*)d_in[7];
    const int*   src_prev = (const int*)  d_in[8];
    const int*   dst_prev = (const int*)  d_in[9];
    const float* W_prev   = (const float*)d_in[10];
    const float* b_prev   = (const float*)d_in[11];

    const float* ef_succ  = (const float*)d_in[12];
    const int*   src_succ = (const int*)  d_in[13];
    const int*   dst_succ = (const int*)  d_in[14];
    const float* W_succ   = (const float*)d_in[15];
    const float* b_succ   = (const float*)d_in[16];

    const float* ef_place  = (const float*)d_in[17];
    const int*   src_place = (const int*)  d_in[18];
    const int*   dst_place = (const int*)  d_in[19];
    const float* W_place   = (const float*)d_in[20];
    const float* b_place   = (const float*)d_in[21];

    const float* ef_serve  = (const float*)d_in[22];
    const int*   src_serve = (const int*)  d_in[23];
    const int*   dst_serve = (const int*)  d_in[24];
    const float* W_serve   = (const float*)d_in[25];
    const float* b_serve   = (const float*)d_in[26];

    float* out = (float*)d_out;
    float* ws  = (float*)d_ws;

    // Workspace layout (floats)
    size_t off = 0;
    float* sum_prev  = ws + off; off += (size_t)N_OP  * OUT_F;
    float* sum_succ  = ws + off; off += (size_t)N_OP  * OUT_F;
    float* sum_serve = ws + off; off += (size_t)N_OP  * OUT_F;
    float* sum_link  = ws + off; off += (size_t)N_DEV * OUT_F;
    float* sum_place = ws + off; off += (size_t)N_DEV * OUT_F;
    float* cnt_prev  = ws + off; off += N_OP;
    float* cnt_succ  = ws + off; off += N_OP;
    float* cnt_serve = ws + off; off += N_OP;
    float* cnt_link  = ws + off; off += N_DEV;
    float* cnt_place = ws + off; off += N_DEV;

    (void)hipMemsetAsync(d_ws, 0, off * sizeof(float), stream);

    const int E_LINK = 80000, E_BIG = 600000;
    const int t_link = E_LINK / 16;      // 5000 tiles
    const int t_big  = E_BIG  / 16;      // 37500 tiles
    const int blk_link = (t_link + TPB_TILES - 1) / TPB_TILES;
    const int blk_big  = (t_big  + TPB_TILES - 1) / TPB_TILES;

    edge_conv_wmma<<<blk_link, 256, 0, stream>>>(
        dev_feats, ef_link, src_link, dst_link, W_link, b_link,
        sum_link, cnt_link, t_link);
    edge_conv_wmma<<<blk_big, 256, 0, stream>>>(
        op_feats, ef_prev, src_prev, dst_prev, W_prev, b_prev,
        sum_prev, cnt_prev, t_big);
    edge_conv_wmma<<<blk_big, 256, 0, stream>>>(
        op_feats, ef_succ, src_succ, dst_succ, W_succ, b_succ,
        sum_succ, cnt_succ, t_big);
    edge_conv_wmma<<<blk_big, 256, 0, stream>>>(
        op_feats, ef_place, src_place, dst_place, W_place, b_place,
        sum_place, cnt_place, t_big);
    edge_conv_wmma<<<blk_big, 256, 0, stream>>>(
        dev_feats, ef_serve, src_serve, dst_serve, W_serve, b_serve,
        sum_serve, cnt_serve, t_big);

    const int n_op_elems  = N_OP  * D_FEAT;
    const int n_dev_elems = N_DEV * D_FEAT;
    finalize3<<<(n_op_elems + 255) / 256, 256, 0, stream>>>(
        op_feats, sum_prev, cnt_prev, sum_succ, cnt_succ, sum_serve, cnt_serve,
        out, N_OP);
    finalize2<<<(n_dev_elems + 255) / 256, 256, 0, stream>>>(
        dev_feats, sum_link, cnt_link, sum_place, cnt_place,
        out + (size_t)N_OP * 256, N_DEV);
}